// UOT_85718957294047
// MI455X (gfx1250) — compile-verified
//
#include <hip/hip_runtime.h>
#include <math.h>

// Problem constants (match reference setup_inputs)
#define BB   64
#define NN   1024
#define DD   512
#define NUM  4
#define EPSV 1e-8f

typedef __attribute__((ext_vector_type(2))) float v2f;
typedef __attribute__((ext_vector_type(8))) float v8f;
typedef __attribute__((ext_vector_type(2))) int   v2i;

// gfx1250 async global->LDS path (guarded; falls back to plain loads)
#if __has_builtin(__builtin_amdgcn_global_load_async_to_lds_b64) && \
    __has_builtin(__builtin_amdgcn_s_wait_asynccnt)
#define UOT_ASYNC 1
typedef __attribute__((address_space(1))) v2i* gv2i_t;   // global v2i*
typedef __attribute__((address_space(3))) v2i* lv2i_t;   // LDS v2i*
#endif

// ds_swizzle xor-swap within wave32 (group-of-32 mode: and=0x1f, xor=mask)
#define SWZ_ADD(v, imm) \
  (v) += __int_as_float(__builtin_amdgcn_ds_swizzle(__float_as_int(v), (imm)))

// ---------------- block reductions (256 threads) ----------------

__device__ __forceinline__ float block_max_256(float v, float* red) {
  const int tid = threadIdx.x;
  red[tid] = v; __syncthreads();
#pragma unroll
  for (int s = 128; s >= 1; s >>= 1) {
    if (tid < s) red[tid] = fmaxf(red[tid], red[tid + s]);
    __syncthreads();
  }
  float r = red[0]; __syncthreads();
  return r;
}

// Sum-reduce 256 values. Tree to 32 partials, then one
// v_wmma_f32_16x16x4_f32 with A=ones, B=(p,0):
//   D[m][j] = sum_k B[k][j] = p_j + p_{j+16}  (exact f32, layout-agnostic
//   since A is all ones), leaving pairsum_{lane%16} in every lane of c[0];
// finish with 4 ds_swizzle xor steps over the 16-index space.
__device__ __forceinline__ float block_sum_256(float v, float* red) {
  const int tid = threadIdx.x;
  red[tid] = v; __syncthreads();
  if (tid < 128) red[tid] += red[tid + 128]; __syncthreads();
  if (tid <  64) red[tid] += red[tid +  64]; __syncthreads();
  if (tid <  32) red[tid] += red[tid +  32]; __syncthreads();
  if (tid < 32) {                // wave 0, EXEC all-ones as WMMA requires
    float p = red[tid];
    v2f a;  a[0] = 1.0f; a[1] = 1.0f;
    v2f bm; bm[0] = p;   bm[1] = 0.0f;
    v8f c = {};
    c = __builtin_amdgcn_wmma_f32_16x16x4_f32(
        /*neg_a=*/false, a, /*neg_b=*/false, bm,
        /*c_mod=*/(short)0, c, /*reuse_a=*/false, /*reuse_b=*/false);
    float ps = c[0];             // pairsum_{lane%16}
    SWZ_ADD(ps, 0x041F);         // SWAPX1
    SWZ_ADD(ps, 0x081F);         // SWAPX2
    SWZ_ADD(ps, 0x101F);         // SWAPX4
    SWZ_ADD(ps, 0x201F);         // SWAPX8
    if (tid == 0) red[0] = ps;
  }
  __syncthreads();
  float r = red[0]; __syncthreads();
  return r;
}

// ---------------- small-state init ----------------
__global__ void uot_init(const float* __restrict__ p0, const float* __restrict__ q0,
                         float* __restrict__ m, float* __restrict__ U,
                         float* __restrict__ e, float* __restrict__ P) {
  const int b = blockIdx.x, tid = threadIdx.x;
  for (int d = tid; d < DD; d += 256) { m[b*DD + d] = logf(p0[b*DD + d]); U[b*DD + d] = 0.0f; }
  for (int n = tid; n < NN; n += 256) { e[b*NN + n] = logf(q0[b*NN + n] + EPSV); P[b*NN + n] = 0.0f; }
}

// ---------------- row pass: update T (normalize over D) ----------------
// one block per (b,n) row; 256 threads * float2 (global_load_b64)
__global__ void uot_row_pass(const float* __restrict__ x,  const float* __restrict__ p0,
                             const float* __restrict__ q0, const float* __restrict__ mask,
                             const float* __restrict__ rhoA,
                             const float* __restrict__ z,  const float* __restrict__ ls,
                             const float* __restrict__ m,  float* __restrict__ out,
                             int k, int first, int finalize) {
  __shared__ float red[256];
  const int tid = threadIdx.x;
  const int row = blockIdx.x;            // b*NN + n
  const int b   = row >> 10;             // NN = 1024
  const size_t base = (size_t)row * DD;
  const float rho = rhoA[k];
  const float ir  = 1.0f / rho;

  const float2 xx = reinterpret_cast<const float2*>(x + base)[tid];
  float y0, y1;
  if (first) {                           // z=0, log_s = log(q0*p0+eps) on the fly
    const float q = q0[row];
    const float2 pp = reinterpret_cast<const float2*>(p0 + b*DD)[tid];
    y0 = xx.x * ir + logf(q * pp.x + EPSV);
    y1 = xx.y * ir + logf(q * pp.y + EPSV);
  } else {
    const float2 zz  = reinterpret_cast<const float2*>(z  + base)[tid];
    const float2 lss = reinterpret_cast<const float2*>(ls + base)[tid];
    y0 = (xx.x - zz.x) * ir + lss.x;
    y1 = (xx.y - zz.y) * ir + lss.y;
  }
  const float M = block_max_256(fmaxf(y0, y1), red);
  const float S = block_sum_256(expf(y0 - M) + expf(y1 - M), red);
  const float L = M + logf(S);
  const float2 mm = reinterpret_cast<const float2*>(m + b*DD)[tid];
  float2 o;
  o.x = mm.x - L + y0;
  o.y = mm.y - L + y1;
  if (finalize) {                        // last iteration: emit exp(log_t)*mask
    const float mk = mask[row];
    o.x = expf(o.x) * mk;
    o.y = expf(o.y) * mk;
  }
  reinterpret_cast<float2*>(out + base)[tid] = o;
}

// ---------------- column pass: update S (normalize over N) + z ----------------
// one block per (b, 32-column tile); threads = 32 (d) x 8 (n-strided).
// e/mask tiles are staged global->LDS asynchronously and overlap phase 1,
// which doesn't need them (they're consumed only in phase 2).
__global__ void uot_col_pass(const float* __restrict__ lt_arr,  // d_out holds log_t
                             const float* __restrict__ mask,
                             const float* __restrict__ rhoA, const float* __restrict__ a1A,
                             float* __restrict__ z, float* __restrict__ ls,
                             const float* __restrict__ e,
                             int k, int first) {
  __shared__ float eL[NN];
  __shared__ float mkL[NN];
  __shared__ float pM[8][33];
  __shared__ float pS[8][33];
  __shared__ float Lsh[32];
  const int tid = threadIdx.x;
  const int tx = tid & 31, ty = tid >> 5;
  const int blk = blockIdx.x;
  const int b = blk >> 4;                       // DD/32 = 16 tiles per batch
  const int d = ((blk & 15) << 5) + tx;
  const float rho = rhoA[k];
  const float inv = 1.0f / (a1A[k] + rho);

#ifdef UOT_ASYNC
  // async global->LDS staging; completion deferred until just before phase 2
#pragma unroll
  for (int i = tid * 2; i < NN; i += 512) {
    __builtin_amdgcn_global_load_async_to_lds_b64(
        (gv2i_t)(e + b*NN + i),    (lv2i_t)&eL[i],  0, 0);
    __builtin_amdgcn_global_load_async_to_lds_b64(
        (gv2i_t)(mask + b*NN + i), (lv2i_t)&mkL[i], 0, 0);
  }
#else
  for (int i = tid; i < NN; i += 256) {
    eL[i]  = e[b*NN + i];
    mkL[i] = mask[b*NN + i];
  }
#endif

  const size_t cb = (size_t)b * NN * DD + d;

  // phase 1: online logsumexp over the column (coalesced along d)
  float M = -INFINITY, S = 0.0f;
  for (int n = ty; n < NN; n += 8) {
    const size_t idx = cb + (size_t)n * DD;
    if (n + 32 < NN) {
      __builtin_prefetch(&lt_arr[idx + (size_t)32*DD], 0, 1);   // global_prefetch_b8
      if (!first) __builtin_prefetch(&z[idx + (size_t)32*DD], 0, 1);
    }
    const float lt = lt_arr[idx];
    const float zv = first ? 0.0f : z[idx];
    const float y2 = (zv + rho * lt) * inv;
    if (y2 > M) { S = S * expf(M - y2) + 1.0f; M = y2; }
    else        { S += expf(y2 - M); }
  }
  pM[ty][tx] = M; pS[ty][tx] = S;
  __syncthreads();   // also orders fallback-path LDS staging stores
  if (ty == 0) {
    float Mc = pM[0][tx], Sc = pS[0][tx];
#pragma unroll
    for (int r = 1; r < 8; ++r) {
      const float M2 = pM[r][tx], S2 = pS[r][tx];
      const float Mn = fmaxf(Mc, M2);
      Sc = Sc * expf(Mc - Mn) + S2 * expf(M2 - Mn);
      Mc = Mn;
    }
    Lsh[tx] = Mc + logf(Sc);
  }
#ifdef UOT_ASYNC
  __builtin_amdgcn_s_wait_asynccnt(0);   // each wave drains its async LDS writes
#endif
  __syncthreads();                       // cross-wave visibility of eL/mkL + Lsh
  const float L = Lsh[tx];

  // phase 2: write log_s and z (phase-1 tile lives in L2: ~256 KB/WG)
  for (int n = ty; n < NN; n += 8) {
    const size_t idx = cb + (size_t)n * DD;
    const float lt = lt_arr[idx];
    const float zv = first ? 0.0f : z[idx];
    const float y2 = (zv + rho * lt) * inv;
    const float lsv = eL[n] - L + y2;
    const float mk = mkL[n];
    const float t = expf(lt)  * mk;
    const float s = expf(lsv) * mk;
    ls[idx] = lsv;
    z[idx]  = zv + rho * (t - s);
  }
}

// ---------------- marginal updates (tiny; separable exact form) ----------------
// log_mu stays a function of d only: alpha_d = (rho*m + a2*log p0 - U)/(rho+a2),
// m' = alpha - LSE_d(alpha), U += rho*exp(m').  (n-parts of z1 cancel in the LSE.)
__global__ void uot_mu(const float* __restrict__ p0, const float* __restrict__ rhoA,
                       const float* __restrict__ a2A,
                       float* __restrict__ m, float* __restrict__ U, int k) {
  __shared__ float red[256];
  const int tid = threadIdx.x, b = blockIdx.x;
  const float rho = rhoA[k], a2 = a2A[k];
  const float inv = 1.0f / (rho + a2);
  const int d0 = b*DD + tid, d1 = d0 + 256;
  const float al0 = (rho*m[d0] + a2*logf(p0[d0]) - U[d0]) * inv;
  const float al1 = (rho*m[d1] + a2*logf(p0[d1]) - U[d1]) * inv;
  const float M = block_max_256(fmaxf(al0, al1), red);
  const float S = block_sum_256(expf(al0 - M) + expf(al1 - M), red);
  const float L = M + logf(S);
  const float m0 = al0 - L, m1 = al1 - L;
  m[d0] = m0; m[d1] = m1;
  U[d0] += rho * expf(m0);
  U[d1] += rho * expf(m1);
}

__global__ void uot_eta(const float* __restrict__ q0, const float* __restrict__ rhoA,
                        const float* __restrict__ a3A,
                        float* __restrict__ e, float* __restrict__ P, int k) {
  __shared__ float red[256];
  const int tid = threadIdx.x, b = blockIdx.x;
  const float rho = rhoA[k], a3 = a3A[k];
  const float inv = 1.0f / (rho + a3);
  float g[4];
  float mx = -INFINITY;
#pragma unroll
  for (int j = 0; j < 4; ++j) {
    const int n = b*NN + tid + 256*j;
    g[j] = (rho*e[n] + a3*logf(q0[n] + EPSV) - P[n]) * inv;
    mx = fmaxf(mx, g[j]);
  }
  const float M = block_max_256(mx, red);
  float sp = 0.0f;
#pragma unroll
  for (int j = 0; j < 4; ++j) sp += expf(g[j] - M);
  const float S = block_sum_256(sp, red);
  const float L = M + logf(S);
#pragma unroll
  for (int j = 0; j < 4; ++j) {
    const int n = b*NN + tid + 256*j;
    const float en = g[j] - L;
    e[n] = en;
    P[n] += rho * expf(en);
  }
}

// ---------------- launch ----------------
extern "C" void kernel_launch(void* const* d_in, const int* in_sizes, int n_in,
                              void* d_out, int out_size, void* d_ws, size_t ws_size,
                              hipStream_t stream) {
  const float* x    = (const float*)d_in[0];
  const float* p0   = (const float*)d_in[1];
  const float* q0   = (const float*)d_in[2];
  const float* a1   = (const float*)d_in[3];
  const float* a2   = (const float*)d_in[4];
  const float* a3   = (const float*)d_in[5];
  const float* rho  = (const float*)d_in[6];
  const float* mask = (const float*)d_in[7];
  float* out = (float*)d_out;

  // workspace: z (134MB) + log_s (134MB) + small marginal state (~0.8MB)
  const size_t BND = (size_t)BB * NN * DD;
  char* ws = (char*)d_ws;
  float* z  = (float*)ws;
  float* ls = (float*)(ws + BND * sizeof(float));
  float* m  = (float*)(ws + 2 * BND * sizeof(float));
  float* U  = m + BB * DD;
  float* e  = U + BB * DD;
  float* P  = e + BB * NN;

  uot_init<<<BB, 256, 0, stream>>>(p0, q0, m, U, e, P);

  for (int k = 0; k < NUM; ++k) {
    const int first = (k == 0);
    const int fin   = (k == NUM - 1);
    uot_row_pass<<<BB * NN, 256, 0, stream>>>(x, p0, q0, mask, rho, z, ls, m, out,
                                              k, first, fin);
    if (k < NUM - 1) {
      uot_col_pass<<<BB * (DD / 32), 256, 0, stream>>>(out, mask, rho, a1, z, ls, e,
                                                       k, first);
      uot_mu<<<BB, 256, 0, stream>>>(p0, rho, a2, m, U, k);   // m needed for k+1 row pass
      if (k < NUM - 2)
        uot_eta<<<BB, 256, 0, stream>>>(q0, rho, a3, e, P, k); // e needed for k+1 col pass
    }
  }
}